// Net_14147622273468
// MI455X (gfx1250) — compile-verified
//
#include <hip/hip_runtime.h>
#include <hip/hip_bf16.h>
#include <math.h>

// ---------------- problem constants (match reference) ----------------
#define N_NODES 100000
#define N_EDGES 3200000
#define F_IN    512
#define H1C     64
#define H2C     32
#define H3C     16
#define NCLS    8
#define N_TILES (N_NODES / 16)   // 6250 exactly

typedef __attribute__((ext_vector_type(2))) float v2f;
typedef __attribute__((ext_vector_type(8))) float v8f;

// ---------------- init: deg=1 (self loop), cnt=0 ----------------
__global__ void init_kernel(float* __restrict__ deg, int* __restrict__ cnt) {
    int i = blockIdx.x * blockDim.x + threadIdx.x;
    if (i < N_NODES) { deg[i] = 1.0f; cnt[i] = 0; }
}

// ---------------- per-edge: in-degree count + weighted degree ----------------
__global__ void edge_count_kernel(int* __restrict__ cnt, float* __restrict__ deg,
                                  const int* __restrict__ col,
                                  const float* __restrict__ w) {
    int e = blockIdx.x * blockDim.x + threadIdx.x;
    if (e < N_EDGES) {
        int c = col[e];
        atomicAdd(&cnt[c], 1);
        atomicAdd(&deg[c], w[e]);
    }
}

__global__ void dinv_kernel(float* __restrict__ deg) {
    int i = blockIdx.x * blockDim.x + threadIdx.x;
    if (i < N_NODES) {
        float d = deg[i];
        deg[i] = (d > 0.0f) ? rsqrtf(d) : 0.0f;   // in place: now holds dinv
    }
}

// ---------------- single-workgroup exclusive scan of cnt -> offs, cursor ----------------
__global__ void scan_kernel(const int* __restrict__ cnt,
                            int* __restrict__ offs, int* __restrict__ cursor) {
    __shared__ int tmp[1024];
    __shared__ int carry;
    const int t = threadIdx.x;
    if (t == 0) carry = 0;
    __syncthreads();
    for (int base = 0; base < N_NODES; base += 1024) {
        int i = base + t;
        int v = (i < N_NODES) ? cnt[i] : 0;
        tmp[t] = v;
        __syncthreads();
        #pragma unroll
        for (int d = 1; d < 1024; d <<= 1) {      // Hillis-Steele inclusive scan in LDS
            int add = (t >= d) ? tmp[t - d] : 0;
            __syncthreads();
            tmp[t] += add;
            __syncthreads();
        }
        int incl = tmp[t];
        int excl = incl - v + carry;
        if (i < N_NODES) { offs[i] = excl; cursor[i] = excl; }
        __syncthreads();
        if (t == 1023) carry += incl;             // chunk total
        __syncthreads();
    }
}

// ---------------- fill CSR edge-id list (by destination) ----------------
__global__ void fill_kernel(const int* __restrict__ col,
                            int* __restrict__ cursor, int* __restrict__ eidx) {
    int e = blockIdx.x * blockDim.x + threadIdx.x;
    if (e < N_EDGES) {
        int pos = atomicAdd(&cursor[col[e]], 1);
        eidx[pos] = e;
    }
}

// ---------------- h = x @ W_conv  (100000x512 @ 512x64), f32 WMMA ----------------
// W_conv staged in LDS as K-pairs: wlds[p*64+n] = {W[2p][n], W[2p+1][n]}  (128 KB).
// A-frag (16x4 f32): lane l -> row M = l%16;  lanes 0-15 hold K=0,1; lanes 16-31 hold K=2,3.
// B-frag (4x16 f32): VGPR v, lanes 0-15 -> K=v, N=lane; lanes 16-31 -> K=v+2, N=lane-16.
// C/D (16x16 f32):   VGPR i, lanes 0-15 -> M=i, N=lane; lanes 16-31 -> M=i+8, N=lane-16.
__global__ void gcn_gemm_kernel(const float* __restrict__ x,
                                const float* __restrict__ Wc,
                                float* __restrict__ h) {
    extern __shared__ char smem[];
    float2* wlds = (float2*)smem;                 // 256 pairs x 64 cols

    // cooperative stage of all of W_conv into paired layout (one-time, coalesced)
    for (int i = threadIdx.x; i < (F_IN / 2) * H1C; i += blockDim.x) {
        int p = i >> 6, n = i & 63;
        float2 pr;
        pr.x = Wc[(size_t)(2 * p) * H1C + n];
        pr.y = Wc[(size_t)(2 * p + 1) * H1C + n];
        wlds[i] = pr;
    }
    __syncthreads();                              // before any wave exits

    const int wave = blockIdx.x * (blockDim.x >> 5) + (threadIdx.x >> 5); // uniform/wave
    if (wave < N_TILES) {
        const int lane  = threadIdx.x & 31;
        const int m     = lane & 15;              // row in tile (A) / col in tile (B,D)
        const int khalf = lane >> 4;              // 0: K=0,1   1: K=2,3

        const float* xrow = x + (size_t)(wave * 16 + m) * F_IN + khalf * 2;

        v8f acc0 = {}, acc1 = {}, acc2 = {}, acc3 = {};

        for (int k = 0; k < F_IN; k += 4) {
            __builtin_prefetch(xrow + k + 32, 0, 0);     // global_prefetch_b8
            v2f a = *(const v2f*)(xrow + k);             // A fragment (global b64)

            const float2* wp = wlds + (((k >> 1) + khalf) << 6) + m;
            v2f b0 = *(const v2f*)(wp + 0);              // ds_load_b64 x4
            v2f b1 = *(const v2f*)(wp + 16);
            v2f b2 = *(const v2f*)(wp + 32);
            v2f b3 = *(const v2f*)(wp + 48);

            acc0 = __builtin_amdgcn_wmma_f32_16x16x4_f32(false, a, false, b0, (short)0, acc0, false, false);
            acc1 = __builtin_amdgcn_wmma_f32_16x16x4_f32(false, a, false, b1, (short)0, acc1, false, false);
            acc2 = __builtin_amdgcn_wmma_f32_16x16x4_f32(false, a, false, b2, (short)0, acc2, false, false);
            acc3 = __builtin_amdgcn_wmma_f32_16x16x4_f32(false, a, false, b3, (short)0, acc3, false, false);
        }

        // store D: row = wave*16 + khalf*8 + i, col = m + 16*tile
        float* hp = h + (size_t)(wave * 16 + khalf * 8) * H1C + m;
        #pragma unroll
        for (int i = 0; i < 8; ++i) {
            hp[(size_t)i * H1C + 0]  = acc0[i];
            hp[(size_t)i * H1C + 16] = acc1[i];
            hp[(size_t)i * H1C + 32] = acc2[i];
            hp[(size_t)i * H1C + 48] = acc3[i];
        }
    }
}

// ---------------- CSR gather-aggregate: zero atomics ----------------
// 64 threads per node (one feature each); edge metadata loads are wave-broadcast,
// h[row] reads are 64 contiguous floats per edge (L2-resident).
__global__ void gather_kernel(const float* __restrict__ h,
                              const float* __restrict__ dinv,
                              const int* __restrict__ row,
                              const float* __restrict__ w,
                              const int* __restrict__ offs,
                              const int* __restrict__ cnt,
                              const int* __restrict__ eidx,
                              float* __restrict__ agg) {
    int tid = blockIdx.x * blockDim.x + threadIdx.x;
    if (tid >= N_NODES * H1C) return;
    int node = tid >> 6;
    int f    = tid & 63;

    float dc = dinv[node];
    float acc = dc * dc * h[(size_t)node * H1C + f];     // self-loop term
    int start = offs[node];
    int end   = start + cnt[node];
    for (int i = start; i < end; ++i) {
        int e = eidx[i];
        int r = row[e];
        float nrm = dinv[r] * w[e] * dc;
        acc = fmaf(nrm, h[(size_t)r * H1C + f], acc);
    }
    agg[(size_t)node * H1C + f] = acc;
}

// ---------------- bias + relu + MLP + log_softmax (one node per thread) ----------------
__global__ void mlp_kernel(const float* __restrict__ agg,
                           const float* __restrict__ bc,
                           const float* __restrict__ W1, const float* __restrict__ b1,
                           const float* __restrict__ W2, const float* __restrict__ b2,
                           const float* __restrict__ W3, const float* __restrict__ b3,
                           float* __restrict__ out) {
    int node = blockIdx.x * blockDim.x + threadIdx.x;
    if (node >= N_NODES) return;

    const float* a = agg + (size_t)node * H1C;

    float h1[H2C];
    #pragma unroll
    for (int j = 0; j < H2C; ++j) h1[j] = b1[j];
    for (int k = 0; k < H1C; ++k) {
        float v = a[k] + bc[k];
        v = v > 0.0f ? v : 0.0f;          // relu of conv output
        const float* wrow = W1 + k * H2C;
        #pragma unroll
        for (int j = 0; j < H2C; ++j) h1[j] = fmaf(v, wrow[j], h1[j]);
    }

    float h2[H3C];
    #pragma unroll
    for (int j = 0; j < H3C; ++j) h2[j] = b2[j];
    #pragma unroll
    for (int k = 0; k < H2C; ++k) {
        float v = h1[k] > 0.0f ? h1[k] : 0.0f;
        const float* wrow = W2 + k * H3C;
        #pragma unroll
        for (int j = 0; j < H3C; ++j) h2[j] = fmaf(v, wrow[j], h2[j]);
    }

    float h3[NCLS];
    #pragma unroll
    for (int j = 0; j < NCLS; ++j) h3[j] = b3[j];
    #pragma unroll
    for (int k = 0; k < H3C; ++k) {
        float v = h2[k] > 0.0f ? h2[k] : 0.0f;
        const float* wrow = W3 + k * NCLS;
        #pragma unroll
        for (int j = 0; j < NCLS; ++j) h3[j] = fmaf(v, wrow[j], h3[j]);
    }

    // log_softmax
    float m = h3[0];
    #pragma unroll
    for (int j = 1; j < NCLS; ++j) m = fmaxf(m, h3[j]);
    float s = 0.0f;
    #pragma unroll
    for (int j = 0; j < NCLS; ++j) s += expf(h3[j] - m);
    float lse = logf(s);
    float* o = out + (size_t)node * NCLS;
    #pragma unroll
    for (int j = 0; j < NCLS; ++j) o[j] = h3[j] - m - lse;
}

// ---------------- launch ----------------
extern "C" void kernel_launch(void* const* d_in, const int* in_sizes, int n_in,
                              void* d_out, int out_size, void* d_ws, size_t ws_size,
                              hipStream_t stream) {
    const float* x   = (const float*)d_in[0];
    const int*   ei  = (const int*)d_in[1];     // [2, N_EDGES]: row = ei, col = ei + N_EDGES
    const float* ew  = (const float*)d_in[2];
    const float* Wc  = (const float*)d_in[3];
    const float* bc  = (const float*)d_in[4];
    const float* W1  = (const float*)d_in[5];
    const float* b1  = (const float*)d_in[6];
    const float* W2  = (const float*)d_in[7];
    const float* b2  = (const float*)d_in[8];
    const float* W3  = (const float*)d_in[9];
    const float* b3  = (const float*)d_in[10];
    float* out = (float*)d_out;

    // workspace carve-up (256B aligned slabs)
    char* ws = (char*)d_ws;
    size_t o = 0;
    auto carve = [&](size_t bytes) { char* p = ws + o; o = (o + bytes + 255) & ~(size_t)255; return p; };
    float* dinv   = (float*)carve((size_t)N_NODES * 4);
    int*   cnt    = (int*)  carve((size_t)N_NODES * 4);
    int*   offs   = (int*)  carve((size_t)N_NODES * 4);
    int*   cursor = (int*)  carve((size_t)N_NODES * 4);
    int*   eidx   = (int*)  carve((size_t)N_EDGES * 4);
    float* h      = (float*)carve((size_t)N_NODES * H1C * 4);
    float* agg    = (float*)carve((size_t)N_NODES * H1C * 4);

    const int* rowp = ei;
    const int* colp = ei + N_EDGES;

    init_kernel<<<(N_NODES + 255) / 256, 256, 0, stream>>>(dinv, cnt);
    edge_count_kernel<<<(N_EDGES + 255) / 256, 256, 0, stream>>>(cnt, dinv, colp, ew);
    dinv_kernel<<<(N_NODES + 255) / 256, 256, 0, stream>>>(dinv);
    scan_kernel<<<1, 1024, 0, stream>>>(cnt, offs, cursor);
    fill_kernel<<<(N_EDGES + 255) / 256, 256, 0, stream>>>(colp, cursor, eidx);

    // 6250 row tiles, 4 waves (128 threads) per block, 128 KB dynamic LDS for W pairs
    gcn_gemm_kernel<<<(N_TILES + 3) / 4, 128, (F_IN / 2) * H1C * sizeof(float2), stream>>>(x, Wc, h);

    gather_kernel<<<(N_NODES * H1C + 255) / 256, 256, 0, stream>>>(h, dinv, rowp, ew, offs, cnt, eidx, agg);

    mlp_kernel<<<(N_NODES + 255) / 256, 256, 0, stream>>>(agg, bc, W1, b1, W2, b2, W3, b3, out);
}